// SSKernelNPLR_52450140619366
// MI455X (gfx1250) — compile-verified
//
#include <hip/hip_runtime.h>

// ---------------------------------------------------------------------------
// SSKernelNPLR on MI455X (gfx1250, wave32).
//   N=64 state, H=256 heads, L=2048, M=L/2+1=1025 freqs.
//   Stage 1: per-(h,n) coefficient precompute (dt folded in).
//   Stage 2: Cauchy sums + Woodbury correction -> real GEMM A-matrix (H x 2052).
//   Stage 3: irfft as f32 WMMA GEMM against an implicit trig basis
//            B[2m][l]=cos(2*pi*m*l/L), B[2m+1][l]=sin(...), realized as a
//            2048-entry interleaved (cos,sin) LDS table indexed by (m*l)&2047.
// ---------------------------------------------------------------------------

#define HH   256
#define NN   64
#define LL   2048
#define MM   1025            // L/2 + 1
#define KPAD 2052            // 2*MM padded to a multiple of 4 (WMMA K-step)

typedef float v2f __attribute__((ext_vector_type(2)));
typedef float v8f __attribute__((ext_vector_type(8)));

// ---------------- Stage 1: per-(h,n) precompute -----------------------------
// pre SoA layout [10][H*N]:
//   0: wr  1: wi          (w = w_complex * dt)
//   2: x00 3: y00         (v_ab = dt * B_a * C_b;  B0=B, B1=P, C0=C, C1=Q)
//   4: x01 5: y01
//   6: x10 7: y10
//   8: x11 9: y11
__global__ void precomp_kernel(const float* __restrict__ C_ri,
                               const float* __restrict__ B_ri,
                               const float* __restrict__ P_ri,
                               const float* __restrict__ Q_ri,
                               const float* __restrict__ w_ri,
                               const float* __restrict__ log_dt,
                               float* __restrict__ pre) {
  int idx = blockIdx.x * blockDim.x + threadIdx.x;   // h*N + n
  if (idx >= HH * NN) return;
  int h = idx >> 6;
  float dt = expf(log_dt[h]);

  float wr = w_ri[2 * idx + 0] * dt;
  float wi = w_ri[2 * idx + 1] * dt;

  float br = B_ri[2 * idx], bi = B_ri[2 * idx + 1];
  float pr = P_ri[2 * idx], pi = P_ri[2 * idx + 1];
  float cr = C_ri[2 * idx], ci = C_ri[2 * idx + 1];
  float qr = Q_ri[2 * idx], qi = Q_ri[2 * idx + 1];

  const int S = HH * NN;
  pre[0 * S + idx] = wr;
  pre[1 * S + idx] = wi;
  // v00 = B*C
  pre[2 * S + idx] = dt * (br * cr - bi * ci);
  pre[3 * S + idx] = dt * (br * ci + bi * cr);
  // v01 = B*Q
  pre[4 * S + idx] = dt * (br * qr - bi * qi);
  pre[5 * S + idx] = dt * (br * qi + bi * qr);
  // v10 = P*C
  pre[6 * S + idx] = dt * (pr * cr - pi * ci);
  pre[7 * S + idx] = dt * (pr * ci + pi * cr);
  // v11 = P*Q
  pre[8 * S + idx] = dt * (pr * qr - pi * qi);
  pre[9 * S + idx] = dt * (pr * qi + pi * qr);
}

// ---------------- Stage 2: Cauchy sums + Woodbury -> A matrix ---------------
// One block per h; threads stride over m in [0, M]. Writes A[h][0..KPAD-1]:
//   A[h][2m]   =  wgt_m * Re(k_f[h,m])
//   A[h][2m+1] = -wgt_m * Im(k_f[h,m])      (pairs with +sin rows of basis)
// wgt folds the irfft normalization: 1/L at m=0 and m=L/2, else 2/L.
__global__ void cauchy_kernel(const float* __restrict__ pre,
                              float* __restrict__ A) {
  __shared__ float sh[10][NN];
  const int h = blockIdx.x;
  const int tid = threadIdx.x;
  for (int i = tid; i < 10 * NN; i += blockDim.x) {
    int arr = i >> 6, n = i & 63;
    sh[arr][n] = pre[arr * (HH * NN) + h * NN + n];
  }
  __syncthreads();

  const float TPOL = 6.28318530717958647692f / (float)LL;   // 2*pi/L
  for (int m = tid; m <= MM; m += blockDim.x) {
    if (m == MM) {  // zero the K padding
      A[h * KPAD + 2 * MM + 0] = 0.f;
      A[h * KPAD + 2 * MM + 1] = 0.f;
      continue;
    }
    float t = TPOL * (float)m;
    float c = cosf(t), s = sinf(t);
    // omega = c - i*s ; z = 2*(1-omega)/(1+omega), full complex division
    // (finite-huge at Nyquist, mirroring the reference; no inf/NaN).
    float na = 2.f * (1.f - c), nb = 2.f * s;   // numerator
    float dc = 1.f + c, dd = -s;                // denominator (1+omega)
    float d2 = dc * dc + dd * dd;
    float zr = (na * dc + nb * dd) / d2;
    float zi = (nb * dc - na * dd) / d2;

    float rr00 = 0.f, ri00 = 0.f, rr01 = 0.f, ri01 = 0.f;
    float rr10 = 0.f, ri10 = 0.f, rr11 = 0.f, ri11 = 0.f;
    for (int n = 0; n < NN; ++n) {
      float wr = sh[0][n], wi = sh[1][n];
      float dr  = zr - wr;
      float di1 = zi - wi;   // z - w
      float di2 = zi + wi;   // z - conj(w)
      float inv1 = 1.f / (dr * dr + di1 * di1);
      float inv2 = 1.f / (dr * dr + di2 * di2);
      float p = dr * inv1, q  = -di1 * inv1;   // 1/(z-w)
      float u = dr * inv2, t2 = -di2 * inv2;   // 1/(z-conj(w))
      // v*inv1 + conj(v)*inv2 = (x*(p+u) - y*(q-t2)) + i*(x*(q+t2) + y*(p-u))
      float s1 = p + u, s2 = q - t2, s3 = q + t2, s4 = p - u;
      float x, y;
      x = sh[2][n]; y = sh[3][n]; rr00 += x * s1 - y * s2; ri00 += x * s3 + y * s4;
      x = sh[4][n]; y = sh[5][n]; rr01 += x * s1 - y * s2; ri01 += x * s3 + y * s4;
      x = sh[6][n]; y = sh[7][n]; rr10 += x * s1 - y * s2; ri10 += x * s3 + y * s4;
      x = sh[8][n]; y = sh[9][n]; rr11 += x * s1 - y * s2; ri11 += x * s3 + y * s4;
    }
    // g = r00 - r01*r10 / (1 + r11)
    float er = 1.f + rr11, ei = ri11;
    float e2 = er * er + ei * ei;
    float nr = rr01 * rr10 - ri01 * ri10;
    float ni = rr01 * ri10 + ri01 * rr10;
    float gr = rr00 - (nr * er + ni * ei) / e2;
    float gi = ri00 - (ni * er - nr * ei) / e2;
    // k_f = g * 2/(1+omega) = g * (2*dc + i*2*s)/d2
    float qr2 = 2.f * dc / d2, qi2 = 2.f * s / d2;
    float kr = gr * qr2 - gi * qi2;
    float ki = gr * qi2 + gi * qr2;
    float wgt = (m == 0 || m == MM - 1) ? (1.f / (float)LL) : (2.f / (float)LL);
    A[h * KPAD + 2 * m + 0] =  wgt * kr;
    A[h * KPAD + 2 * m + 1] = -wgt * ki;
  }
}

// ---------------- Stage 3: irfft as f32 WMMA GEMM ---------------------------
// out[h][l] = sum_k A[h][k] * Bmat[k][l], Bmat rows = interleaved cos/sin.
// Grid (16,16): blockIdx.x -> 16-row tile, blockIdx.y -> 128-col group.
// 256 threads = 8 wave32s; each wave owns one 16x16 output tile and loops
// K in steps of 4 with V_WMMA_F32_16X16X4_F32.
//
// A fragment (ISA 16x4 f32 layout): VGPR0 = K=0 (lanes 0-15) / K=2 (16-31);
//                                   VGPR1 = K=1 / K=3. -> per-lane float2 load.
// B fragment (4x16, rows striped across lanes per VGPR, mirroring A):
//   VGPR0 = rows K=0/K=2, VGPR1 = rows K=1/K=3. With K-base 4j, each lane's
//   two entries are (cos, sin) of the same m = 2j + (lane>=16), same column
//   -> one ds_load_b64 from the interleaved trig table at (m*l) & 2047,
//   index maintained by an add-and-mask recurrence (no per-step multiply).
__global__ void __launch_bounds__(256)
irfft_wmma_kernel(const float* __restrict__ A, float* __restrict__ out) {
  __shared__ v2f tab[LL];   // 16 KB: tab[i] = (cos(2*pi*i/L), sin(2*pi*i/L))
  const int tid = threadIdx.x;
  for (int i = tid; i < LL; i += 256) {
    float ang = (6.28318530717958647692f / (float)LL) * (float)i;
    v2f v; v.x = cosf(ang); v.y = sinf(ang);
    tab[i] = v;
  }
  __syncthreads();

  const int lane = tid & 31;
  const int wv   = tid >> 5;
  const int r16  = lane & 15;
  const int half = lane >> 4;                 // 0: K+0/K+1, 1: K+2/K+3
  const int h0   = blockIdx.x * 16;
  const int col  = blockIdx.y * 128 + wv * 16 + r16;

  const float* aptr = A + (h0 + r16) * KPAD + 2 * half;
  unsigned idx  = ((unsigned)(half * col)) & (LL - 1);   // (m*col) & 2047, m=half at j=0
  unsigned step = ((unsigned)(2 * col)) & (LL - 1);      // m += 2 per K-step

  v8f acc = {0.f, 0.f, 0.f, 0.f, 0.f, 0.f, 0.f, 0.f};
  for (int j = 0; j < KPAD / 4; ++j) {
    v2f afrag = *(const v2f*)(aptr);   // A[row][4j+2*half .. +1]  (8B aligned)
    aptr += 4;
    v2f bfrag = tab[idx];              // (cos, sin) for m = 2j+half at this col
    idx = (idx + step) & (LL - 1);
    acc = __builtin_amdgcn_wmma_f32_16x16x4_f32(
        /*neg_a=*/false, afrag, /*neg_b=*/false, bfrag,
        /*c_mod=*/(short)0, acc, /*reuse_a=*/false, /*reuse_b=*/false);
  }

  // C/D layout: VGPR r -> M = r (lanes 0-15) or 8+r (lanes 16-31), N = lane&15.
  float* op = out + (h0 + 8 * half) * LL + col;
#pragma unroll
  for (int r = 0; r < 8; ++r) op[r * LL] = acc[r];
}

// ---------------------------------------------------------------------------
extern "C" void kernel_launch(void* const* d_in, const int* in_sizes, int n_in,
                              void* d_out, int out_size, void* d_ws, size_t ws_size,
                              hipStream_t stream) {
  const float* C_ri  = (const float*)d_in[0];
  const float* B_ri  = (const float*)d_in[1];
  const float* P_ri  = (const float*)d_in[2];
  const float* Q_ri  = (const float*)d_in[3];
  const float* w_ri  = (const float*)d_in[4];
  const float* logdt = (const float*)d_in[5];
  (void)in_sizes; (void)n_in; (void)out_size; (void)ws_size;

  float* ws   = (float*)d_ws;
  float* pre  = ws;                      // 10 * H * N floats      (640 KB)
  float* Amat = ws + 10 * HH * NN;       // H * KPAD floats        (2.1 MB)

  precomp_kernel<<<(HH * NN + 255) / 256, 256, 0, stream>>>(
      C_ri, B_ri, P_ri, Q_ri, w_ri, logdt, pre);
  cauchy_kernel<<<HH, 256, 0, stream>>>(pre, Amat);
  irfft_wmma_kernel<<<dim3(16, 16), 256, 0, stream>>>(Amat, (float*)d_out);
}